// prms_marrmot_changed_2250562863374
// MI455X (gfx1250) — compile-verified
//
#include <hip/hip_runtime.h>
#include <hip/hip_bf16.h>
#include <stdint.h>

// ---------------------------------------------------------------------------
// PRMS/MARRMoT bucket model, CDNA5 (gfx1250).
//
// B=2000 basins x NMUL=4 = 8000 independent trajectories, D=365 sequential
// steps, 25 fluxes/step out. Output (365,2000,25,4) f32 = 292 MB -> purely
// store-bound (~12.5 us at 23.3 TB/s). No matmul -> no WMMA; the CDNA5
// feature that fits is the Tensor Data Mover.
//
// Partitioning: 250 single-wave workgroups (32 threads = 8 basins x 4 mult),
// exact fit, no tail. Single-wave WGs need no barriers (barrier ops are
// S_NOP per ISA 3.1), so each wave privately double-buffers a contiguous
// 3.2 KB day-slice in LDS and fires one TENSOR_STORE_FROM_LDS per day:
//   compute day d -> s_wait_tensorcnt(1)  (TDM d-2 retired, buffer free)
//   ds_store 25 fluxes/lane              (bank-conflict-free, 100 mod 64)
//   s_wait_dscnt 0                        (staging visible to TDM engine)
//   tensor_store_from_lds                 (3.2 KB contiguous DMA to HBM)
// VALU never waits on STOREcnt; the DMA engines emit the 292 MB.
// ---------------------------------------------------------------------------

#define NPARAM 21
#define NFLUX  25
#define NMUL   4

static constexpr int   Bc  = 2000;
static constexpr int   Dc  = 365;
static constexpr int   BASINS_PER_BLOCK = 8;
static constexpr int   THREADS      = BASINS_PER_BLOCK * NMUL;            // 32 = 1 wave
static constexpr int   STAGE_FLOATS = BASINS_PER_BLOCK * NFLUX * NMUL;    // 800 -> 3.2 KB
static constexpr float EPSc = 1e-9f;

typedef __attribute__((ext_vector_type(4))) unsigned int u32x4;
typedef __attribute__((ext_vector_type(8))) int          i32x8;
typedef __attribute__((ext_vector_type(4))) int          i32x4;

__constant__ float c_LO[NPARAM] = {
    -3.0f, 0.0f, 0.0f, 0.0f, 1.0f, 1.0f, 0.05f, 0.05f, 0.01f, 0.005f,
     1.0f, 1.0f, 1.0f, 0.0f, 1.0f, 0.01f, 1.0f, 0.01f, 0.001f, 0.01f, 0.001f};
__constant__ float c_HI[NPARAM] = {
     3.0f, 20.0f, 1.0f, 1.0f, 5.0f, 50.0f, 0.95f, 0.95f, 0.9f, 0.995f,
     2000.0f, 500.0f, 1000.0f, 20.0f, 300.0f, 1.0f, 5.0f, 1.0f, 0.1f, 1.0f, 0.1f};

__device__ __forceinline__ float sigm(float x) { return 1.0f / (1.0f + __expf(x)); }

// _smooth_T with r = 0.01
__device__ __forceinline__ float smooth_T(float Tv, float Tt) {
    return sigm((Tv - Tt) * 100.0f);
}

// _smooth_S with e = 1, generic r (reference default r = 2)
__device__ __forceinline__ float smooth_S(float S, float Smax, float r) {
    Smax = fmaxf(Smax, 0.0f);
    float num    = S - Smax + r * Smax;
    float den    = r * Smax;
    float safe_r = (r == 0.0f) ? 1.0f : r;
    float arg    = (den == 0.0f) ? (num / safe_r) : (num / den);
    return sigm(arg);
}

__global__ __launch_bounds__(THREADS)
void prms_tdm_kernel(const float* __restrict__ P,
                     const float* __restrict__ Ep,
                     const float* __restrict__ T,
                     const float* __restrict__ praw,
                     const float* __restrict__ S_init,
                     float* __restrict__ out)
{
    __shared__ float stage[2][STAGE_FLOATS];   // wave-private double buffer

    const int tid = threadIdx.x;               // 0..31, exactly one wave
    const int m   = tid & (NMUL - 1);
    const int bl  = tid >> 2;                  // basin within block (0..7)
    const int b0  = blockIdx.x * BASINS_PER_BLOCK;
    const int b   = b0 + bl;                   // 250*8 == 2000: always in range

    // ---- load + rescale parameters, load initial states -------------------
    float pv[NPARAM];
#pragma unroll
    for (int i = 0; i < NPARAM; ++i)
        pv[i] = praw[((size_t)b * NPARAM + i) * NMUL + m] * (c_HI[i] - c_LO[i]) + c_LO[i];
    const float tt = pv[0],  ddf  = pv[1],  alpha = pv[2],  beta   = pv[3];
    const float stor = pv[4], retip = pv[5]; /* fscn=pv[6] unused */
    const float scx = pv[7],  scn  = pv[8]; /* flz=pv[9], stot=pv[10] unused */
    const float remx = pv[11], smax = pv[12], cgw = pv[13], resmax = pv[14];
    const float k1 = pv[15], k2 = pv[16], k3 = pv[17], k4 = pv[18], k5 = pv[19], k6 = pv[20];

    float S1 = S_init[((size_t)b*7 + 0)*NMUL + m];
    float S2 = S_init[((size_t)b*7 + 1)*NMUL + m];
    float S3 = S_init[((size_t)b*7 + 2)*NMUL + m];
    float S4 = S_init[((size_t)b*7 + 3)*NMUL + m];
    float S5 = S_init[((size_t)b*7 + 4)*NMUL + m];
    float S6 = S_init[((size_t)b*7 + 5)*NMUL + m];
    float S7 = S_init[((size_t)b*7 + 6)*NMUL + m];

    const float* Prow  = P  + (size_t)b * Dc;
    const float* Eprow = Ep + (size_t)b * Dc;
    const float* Trow  = T  + (size_t)b * Dc;

    float* const stA = &stage[0][bl * (NFLUX * NMUL) + m];
    float* const stB = &stage[1][bl * (NFLUX * NMUL) + m];

    // ---- time loop --------------------------------------------------------
    for (int d = 0; d < Dc; ++d) {
        const float Pd  = Prow[d];
        const float Epd = Eprow[d];
        const float Td  = Trow[d];

        const float sT   = smooth_T(Td, tt);
        const float ps   = Pd * sT;
        const float pr   = Pd * (1.0f - sT);
        const float pim  = (1.0f - beta) * pr;
        const float psm  = beta * pr;
        const float pby  = (1.0f - alpha) * psm;
        const float pin  = alpha * psm;
        const float ptf  = pin * (1.0f - smooth_S(S2, stor, 2.0f));
        const float m_   = fmaxf(fminf(ddf * (Td - tt), S1), 0.0f);     // dt = 1
        const float mim  = (1.0f - beta) * m_;
        const float msm  = beta * m_;
        const float sas  = (pim + mim) * (1.0f - smooth_S(S3, retip, 2.0f));
        const float to_soil = msm + ptf + pby;
        const float sro  = (scn + (scx - scn) * S4 / remx) * to_soil;
        const float inf_ = fmaxf(to_soil - sro, 0.0f);
        const float pc   = inf_ * (1.0f - smooth_S(S4, remx, 2.0f));
        const float excs = pc * (1.0f - smooth_S(S5, smax, 2.0f));
        const float sep  = fminf(cgw, excs);
        const float qres = fmaxf(excs - sep, 0.0f);
        const float base6= fmaxf(S6, 0.0f);
        const float gad  = k1 * __powf(base6 / resmax + EPSc, k2);
        const float ras  = fminf(base6, k3 * base6 + k4 * base6 * base6);
        const float bas  = k5 * S7;
        const float snk  = k6 * S7;
        const float ein  = fminf(S2, beta * Epd);
        const float eim  = fminf(S3, (1.0f - beta) * Epd);
        const float ep1  = fmaxf(Epd - ein - eim, 0.0f);
        const float ea   = fminf(S4 / remx * ep1, S4);
        const float ep2  = fmaxf(Epd - ein - eim - ea, 0.0f);
        const float et   = fmaxf(S5 / smax * ep2 *
                                 smooth_S(S4, Epd - ein - eim, S5), 0.0f);  // r = S5/dt

        float fr[NFLUX];
        fr[0]=ps;  fr[1]=pr;  fr[2]=pim;  fr[3]=psm;  fr[4]=pby;
        fr[5]=pin; fr[6]=ptf; fr[7]=m_;   fr[8]=mim;  fr[9]=msm;
        fr[10]=sas; fr[11]=sro; fr[12]=inf_; fr[13]=pc; fr[14]=excs;
        fr[15]=qres; fr[16]=sep; fr[17]=gad; fr[18]=ras; fr[19]=bas;
        fr[20]=snk; fr[21]=ein; fr[22]=eim; fr[23]=ea; fr[24]=et;

        S1 = fmaxf(S1 + ps - m_, 0.0f);
        S2 = fmaxf(S2 + pin - ein - ptf, 0.0f);
        S3 = fmaxf(S3 + pim + mim - eim - sas, 0.0f);
        S4 = fmaxf(S4 + inf_ - ea - pc, 0.0f);
        S5 = fmaxf(S5 + pc - et - excs, 0.0f);
        S6 = fmaxf(S6 + qres - gad - ras, 0.0f);
        S7 = fmaxf(S7 + sep + gad - bas - snk, 0.0f);

        // Buffer d&1 was last consumed by TDM(d-2); this wave's TDM ops
        // retire in order, so tensorcnt<=1 guarantees it is free.
        __builtin_amdgcn_s_wait_tensorcnt(1);

        float* const st = (d & 1) ? stB : stA;
#pragma unroll
        for (int f = 0; f < NFLUX; ++f)
            st[f * NMUL] = fr[f];              // 25 ds_store_b32, conflict-free

        // Staging must be in LDS before the TDM engine reads it.
        asm volatile("s_wait_dscnt 0x0" ::: "memory");

        // One contiguous 3.2 KB DMA: LDS -> out[(d*B + b0)*100 ...]
        {
            const uint64_t ga = (uint64_t)(uintptr_t)out +
                                ((uint64_t)((size_t)d * Bc + b0) * (NFLUX * NMUL)) * sizeof(float);
            const unsigned la = (unsigned)(uintptr_t)(&stage[d & 1][0]);   // LDS byte offset
            const unsigned n  = (unsigned)STAGE_FLOATS;                    // 800 dwords

            u32x4 g0;
            g0.x = 1u;                                              // count=1, user descriptor
            g0.y = la;                                              // lds_addr
            g0.z = (unsigned)(ga & 0xFFFFFFFFull);                  // global_addr[31:0]
            g0.w = (unsigned)((ga >> 32) & 0x01FFFFFFull)           // global_addr[56:32]
                 | (2u << 30);                                      // type = 2 ("image")

            i32x8 g1;
            g1[0] = (int)(2u << 16);                                // data_size=4B, mask=0, flags=0
            g1[1] = (int)((n & 0xFFFFu) << 16);                     // tensor_dim0[15:0]
            g1[2] = (int)((n >> 16) & 0xFFFFu) | (1 << 16);         // tensor_dim0[31:16] | tensor_dim1=1
            g1[3] = (int)((n & 0xFFFFu) << 16);                     // tile_dim0 = 800
            g1[4] = 0;                                              // tile_dim1=0, tile_dim2=0 (1-D)
            g1[5] = (int)n;                                         // tensor_dim0_stride lo32
            g1[6] = 0;
            g1[7] = 0;

            i32x4 gz  = (i32x4)0;                                   // groups 2/3 unused (<=2-D)
            i32x8 gz8 = (i32x8)0;
            __builtin_amdgcn_tensor_store_from_lds(g0, g1, gz, gz, gz8, 0);
        }
    }

    __builtin_amdgcn_s_wait_tensorcnt(0);      // drain before endpgm
}

extern "C" void kernel_launch(void* const* d_in, const int* in_sizes, int n_in,
                              void* d_out, int out_size, void* d_ws, size_t ws_size,
                              hipStream_t stream) {
    const float* P    = (const float*)d_in[0];
    const float* Ep   = (const float*)d_in[1];
    const float* T    = (const float*)d_in[2];
    const float* praw = (const float*)d_in[3];
    const float* Sini = (const float*)d_in[4];
    float*       out  = (float*)d_out;

    const int nblocks = Bc / BASINS_PER_BLOCK;   // 250, exact
    prms_tdm_kernel<<<dim3(nblocks), dim3(THREADS), 0, stream>>>(P, Ep, T, praw, Sini, out);
}